// LGMRec_1915555414101
// MI455X (gfx1250) — compile-verified
//
#include <hip/hip_runtime.h>
#include <math.h>

#define N_USERS 30000
#define N_ITEMS 30000
#define N_NODES (N_USERS + N_ITEMS)
#define D 64
#define IMG_DIM 2048
#define TXT_DIM 384
#define CDIV(a, b) (((a) + (b) - 1) / (b))

typedef __attribute__((ext_vector_type(2))) float v2f;
typedef __attribute__((ext_vector_type(8))) float v8f;

// ---------------------------------------------------------------------------
// SpMM scatter: out[row] += val * x[col]   (row-major, D=64 columns)
// 64 threads per nnz, one float per thread; unsafeAtomicAdd guarantees the
// hardware GLOBAL_ATOMIC_ADD_F32 path (no CAS retry loop).
// ---------------------------------------------------------------------------
__global__ __launch_bounds__(256) void spmm_scatter(
    const int* __restrict__ rows, const int* __restrict__ cols,
    const float* __restrict__ vals, const float* __restrict__ x,
    float* __restrict__ out, int nnz) {
  long long t = (long long)blockIdx.x * blockDim.x + threadIdx.x;
  long long e = t >> 6;
  if (e >= nnz) return;
  int d = (int)(t & 63);
  int r = rows[e];
  int c = cols[e];
  float v = vals[e];
  float contrib = v * x[(long long)c * D + d];
  unsafeAtomicAdd(&out[(long long)r * D + d], contrib);
}

// ---------------------------------------------------------------------------
// WMMA f32 GEMM: C[M x 64] = A[M x K] * B[K x 64], all row-major f32.
// Block = 128 threads (4 waves). Block computes 64 rows; each wave a 16-row
// slab split into 4 N-tiles of 16. K staged through LDS in chunks of 64.
//
// A tile: row-major, rows padded to 68 floats (keeps 8B row alignment, and
//   spreads the stride-68 fragment reads across all 64 LDS banks).
// B tile: K-pair interleaved -> lB2[(k>>1)*128 + n*2 + (k&1)], so a WMMA B
//   fragment (B[k][n], B[k+1][n]) is one aligned ds_load_b64.
// ---------------------------------------------------------------------------
#define LDP 68

__global__ __launch_bounds__(128) void gemm_wmma_f32(
    const float* __restrict__ A, const float* __restrict__ B,
    float* __restrict__ C, int M, int K) {
  __shared__ float lA[64 * LDP];
  __shared__ float lB2[32 * 128];
  int wave = threadIdx.x >> 5;
  int lane = threadIdx.x & 31;
  int rowBase = blockIdx.x * 64;

  v8f acc[4] = {v8f{}, v8f{}, v8f{}, v8f{}};

  for (int k0 = 0; k0 < K; k0 += 64) {
    __syncthreads();
    // Stage A[rowBase:rowBase+64, k0:k0+64] row-major (float4 I/O).
    for (int i = threadIdx.x; i < 64 * 16; i += 128) {
      int r = i >> 4, c4 = i & 15;
      int gr = rowBase + r;
      if (gr >= M) gr = M - 1;  // clamp (stores guarded later)
      float4 av = ((const float4*)(A + (long long)gr * K + k0))[c4];
      ((float4*)(lA + r * LDP))[c4] = av;
    }
    // Stage B[k0:k0+64, 0:64] pair-interleaved: each i covers a (row-pair,
    // col-pair) cell -> one float4 LDS store {B[2rp][c],B[2rp+1][c],
    // B[2rp][c+1],B[2rp+1][c+1]}.
    for (int i = threadIdx.x; i < 32 * 32; i += 128) {
      int rp = i >> 5;  // 0..31 row pair
      int c2 = i & 31;  // 0..31 column pair
      const float* g = B + (long long)(k0 + 2 * rp) * D + 2 * c2;
      float2 b0 = *(const float2*)g;
      float2 b1 = *(const float2*)(g + D);
      float4 st = make_float4(b0.x, b1.x, b0.y, b1.y);
      ((float4*)(lB2 + rp * 128))[c2] = st;
    }
    __syncthreads();

    int m16 = lane & 15;
    int khalf = (lane >> 4) * 2;  // lanes 0-15: K pair {0,1}; 16-31: {2,3}
    int arow = wave * 16 + m16;
#pragma unroll 4
    for (int kk = 0; kk < 64; kk += 4) {
      v2f a = *(const v2f*)(lA + arow * LDP + kk + khalf);
#pragma unroll
      for (int nt = 0; nt < 4; ++nt) {
        v2f b = *(const v2f*)(lB2 + ((kk + khalf) >> 1) * 128 +
                              (nt * 16 + m16) * 2);
#if __has_builtin(__builtin_amdgcn_wmma_f32_16x16x4_f32)
        acc[nt] = __builtin_amdgcn_wmma_f32_16x16x4_f32(
            false, a, false, b, (short)0, acc[nt], false, false);
#else
#pragma unroll
        for (int r = 0; r < 8; ++r) acc[nt][r] += a.x * b.x + a.y * b.y;
#endif
      }
    }
  }

  // Store D: VGPR r holds M = r (lanes 0-15) / r+8 (lanes 16-31), N = lane&15.
  int mhi = (lane >> 4) * 8;
  int n0 = lane & 15;
#pragma unroll
  for (int r = 0; r < 8; ++r) {
    int grow = rowBase + wave * 16 + mhi + r;
    if (grow < M) {
#pragma unroll
      for (int nt = 0; nt < 4; ++nt)
        C[(long long)grow * D + nt * 16 + n0] = acc[nt][r];
    }
  }
}

// ---------------------------------------------------------------------------
// Elementwise kernels
// ---------------------------------------------------------------------------
__global__ __launch_bounds__(256) void gate_common_k(
    const float* __restrict__ f, const float* __restrict__ fg,
    const float* __restrict__ bias, float* __restrict__ com,
    float* __restrict__ fmc, long long n) {
  long long t = (long long)blockIdx.x * blockDim.x + threadIdx.x;
  if (t >= n) return;
  int d = (int)(t & 63);
  float s = 1.0f / (1.0f + expf(-(fg[t] + bias[d])));
  float c = s * f[t];
  com[t] = c;
  fmc[t] = f[t] - c;
}

__global__ __launch_bounds__(256) void gate_personal_k(
    const float* __restrict__ f, const float* __restrict__ pg,
    const float* __restrict__ bias1, float* __restrict__ pers, long long n) {
  long long t = (long long)blockIdx.x * blockDim.x + threadIdx.x;
  if (t >= n) return;
  int d = (int)(t & 63);
  float s = 1.0f / (1.0f + expf(-(pg[t] + bias1[d])));
  pers[t] = s * f[t];
}

__global__ __launch_bounds__(256) void accumulate_k(
    float* __restrict__ acc, const float* __restrict__ x, long long n) {
  long long t = (long long)blockIdx.x * blockDim.x + threadIdx.x;
  if (t < n) acc[t] += x[t];
}

__global__ __launch_bounds__(256) void scale_rows_k(
    float* __restrict__ x, const float* __restrict__ s, long long n) {
  long long t = (long long)blockIdx.x * blockDim.x + threadIdx.x;
  if (t < n) x[t] *= s[t >> 6];
}

// ---------------------------------------------------------------------------
// Final combine: out = cge/3 + 0.3*l2n(w0*pev+w1*pet) + 0.8*l2n(w0*cev+w1*cet)
// One wave32 per row (2 floats/lane), shfl_xor reduction for the norms.
// ---------------------------------------------------------------------------
__global__ __launch_bounds__(256) void final_combine_k(
    const float* __restrict__ cge, const float* __restrict__ pev,
    const float* __restrict__ pet, const float* __restrict__ cev,
    const float* __restrict__ cet, const float* __restrict__ att,
    float* __restrict__ out, int n_rows) {
  int wave = threadIdx.x >> 5;
  int lane = threadIdx.x & 31;
  int row = blockIdx.x * 8 + wave;
  if (row >= n_rows) return;
  float a0 = att[0], a1 = att[1];
  float mx = fmaxf(a0, a1);
  float e0 = expf(a0 - mx), e1 = expf(a1 - mx);
  float w0 = e0 / (e0 + e1), w1 = e1 / (e0 + e1);
  long long base = (long long)row * D;
  float mg[2], cm[2];
  float ssm = 0.f, ssc = 0.f;
#pragma unroll
  for (int j = 0; j < 2; ++j) {
    long long o = base + lane + j * 32;
    float m = w0 * pev[o] + w1 * pet[o];
    float c = w0 * cev[o] + w1 * cet[o];
    mg[j] = m; cm[j] = c;
    ssm += m * m; ssc += c * c;
  }
#pragma unroll
  for (int off = 16; off > 0; off >>= 1) {
    ssm += __shfl_xor(ssm, off, 32);
    ssc += __shfl_xor(ssc, off, 32);
  }
  float inm = 1.0f / (sqrtf(ssm) + 1e-12f);
  float inc = 1.0f / (sqrtf(ssc) + 1e-12f);
#pragma unroll
  for (int j = 0; j < 2; ++j) {
    long long o = base + lane + j * 32;
    out[o] = cge[o] * (1.0f / 3.0f) + 0.3f * mg[j] * inm + 0.8f * cm[j] * inc;
  }
}

// ---------------------------------------------------------------------------
// Host-side orchestration
// ---------------------------------------------------------------------------
static inline void run_spmm(const int* idx, const float* vals, const float* x,
                            float* out, int nnz, hipStream_t s) {
  long long total = (long long)nnz * D;
  unsigned blocks = (unsigned)CDIV(total, 256);
  spmm_scatter<<<blocks, 256, 0, s>>>(idx, idx + nnz, vals, x, out, nnz);
}

static inline void run_gemm(const float* A, const float* B, float* C, int M,
                            int K, hipStream_t s) {
  gemm_wmma_f32<<<CDIV(M, 64), 128, 0, s>>>(A, B, C, M, K);
}

extern "C" void kernel_launch(void* const* d_in, const int* in_sizes, int n_in,
                              void* d_out, int out_size, void* d_ws,
                              size_t ws_size, hipStream_t stream) {
  const float* user_emb    = (const float*)d_in[0];
  const float* item_emb    = (const float*)d_in[1];
  const float* image_feats = (const float*)d_in[2];
  const float* text_feats  = (const float*)d_in[3];
  const float* trs_v   = (const float*)d_in[4];
  const float* trs_t   = (const float*)d_in[5];
  const float* gate_v  = (const float*)d_in[6];
  const float* gate1_v = (const float*)d_in[7];
  const float* gate_t  = (const float*)d_in[8];
  const float* gate1_t = (const float*)d_in[9];
  const float* bias_v  = (const float*)d_in[10];
  const float* bias1_v = (const float*)d_in[11];
  const float* bias_t  = (const float*)d_in[12];
  const float* bias1_t = (const float*)d_in[13];
  const float* att        = (const float*)d_in[14];
  const float* num_inters = (const float*)d_in[15];
  const float* vals_norm   = (const float*)d_in[16];
  const float* vals_adj    = (const float*)d_in[17];
  const float* vals_img_ii = (const float*)d_in[18];
  const float* vals_txt_ii = (const float*)d_in[19];
  const float* vals_img_uu = (const float*)d_in[20];
  const float* vals_txt_uu = (const float*)d_in[21];
  const float* vals_cii    = (const float*)d_in[22];
  const float* vals_cuu    = (const float*)d_in[23];
  const int* idx_norm   = (const int*)d_in[24];
  const int* idx_adj    = (const int*)d_in[25];
  const int* idx_img_ii = (const int*)d_in[26];
  const int* idx_txt_ii = (const int*)d_in[27];
  const int* idx_img_uu = (const int*)d_in[28];
  const int* idx_txt_uu = (const int*)d_in[29];
  const int* idx_cii    = (const int*)d_in[30];
  const int* idx_cuu    = (const int*)d_in[31];

  const int nnz_norm = in_sizes[16];
  const int nnz_adj  = in_sizes[17];
  const int nnz_g    = in_sizes[18];

  const long long NE = (long long)N_NODES * D;  // 60000*64
  const long long IE = (long long)N_ITEMS * D;  // 30000*64
  const size_t NEb = NE * sizeof(float);

  // Workspace layout (floats)
  float* ws = (float*)d_ws;
  size_t o = 0;
  float* EGO_A = ws + o; o += NE;
  float* EGO_B = ws + o; o += NE;
  float* CGE   = ws + o; o += NE;
  float* F     = ws + o; o += IE;
  float* TMP   = ws + o; o += IE;
  float* FMC   = ws + o; o += IE;
  float* COM_V  = ws + o; o += IE;
  float* PERS_V = ws + o; o += IE;
  float* COM_T  = ws + o; o += IE;
  float* PERS_T = ws + o; o += IE;
  float* UTMP   = ws + o; o += IE;
  float* PERS_EMB_V = ws + o; o += NE;
  float* PERS_EMB_T = ws + o; o += NE;
  float* COM_EMB_V  = ws + o; o += NE;
  float* COM_EMB_T  = ws + o; o += NE;

  const long long UE = (long long)N_USERS * D;
  unsigned ie_blocks = (unsigned)CDIV(IE, 256);
  unsigned ne_blocks = (unsigned)CDIV(NE, 256);
  unsigned ue_blocks = (unsigned)CDIV(UE, 256);

  // ---- Collaborative graph embedding (3-layer LightGCN mean) ----
  hipMemcpyAsync(EGO_A, user_emb, UE * sizeof(float),
                 hipMemcpyDeviceToDevice, stream);
  hipMemcpyAsync(EGO_A + UE, item_emb, IE * sizeof(float),
                 hipMemcpyDeviceToDevice, stream);
  hipMemsetAsync(CGE, 0, NEb, stream);
  float* src = EGO_A;
  float* dst = EGO_B;
  for (int l = 0; l < 3; ++l) {
    hipMemsetAsync(dst, 0, NEb, stream);
    run_spmm(idx_norm, vals_norm, src, dst, nnz_norm, stream);
    accumulate_k<<<ne_blocks, 256, 0, stream>>>(CGE, dst, NE);
    float* t = src; src = dst; dst = t;
  }

  // ---- Modality gating (v then t) ----
  run_gemm(image_feats, trs_v, F, N_ITEMS, IMG_DIM, stream);
  run_gemm(F, gate_v, TMP, N_ITEMS, D, stream);
  gate_common_k<<<ie_blocks, 256, 0, stream>>>(F, TMP, bias_v, COM_V, FMC, IE);
  run_gemm(FMC, gate1_v, TMP, N_ITEMS, D, stream);
  gate_personal_k<<<ie_blocks, 256, 0, stream>>>(F, TMP, bias1_v, PERS_V, IE);

  run_gemm(text_feats, trs_t, F, N_ITEMS, TXT_DIM, stream);
  run_gemm(F, gate_t, TMP, N_ITEMS, D, stream);
  gate_common_k<<<ie_blocks, 256, 0, stream>>>(F, TMP, bias_t, COM_T, FMC, IE);
  run_gemm(FMC, gate1_t, TMP, N_ITEMS, D, stream);
  gate_personal_k<<<ie_blocks, 256, 0, stream>>>(F, TMP, bias1_t, PERS_T, IE);

  // ---- Modal graph propagation (personal uses per-modality ii/uu,
  //      common uses shared cii/cuu) ----
  // v, personal:
  hipMemsetAsync(UTMP, 0, UE * sizeof(float), stream);
  run_spmm(idx_adj, vals_adj, PERS_V, UTMP, nnz_adj, stream);
  scale_rows_k<<<ue_blocks, 256, 0, stream>>>(UTMP, num_inters, UE);
  hipMemsetAsync(PERS_EMB_V, 0, NEb, stream);
  run_spmm(idx_img_uu, vals_img_uu, UTMP, PERS_EMB_V, nnz_g, stream);
  run_spmm(idx_img_ii, vals_img_ii, PERS_V, PERS_EMB_V + UE, nnz_g, stream);
  // v, common:
  hipMemsetAsync(UTMP, 0, UE * sizeof(float), stream);
  run_spmm(idx_adj, vals_adj, COM_V, UTMP, nnz_adj, stream);
  scale_rows_k<<<ue_blocks, 256, 0, stream>>>(UTMP, num_inters, UE);
  hipMemsetAsync(COM_EMB_V, 0, NEb, stream);
  run_spmm(idx_cuu, vals_cuu, UTMP, COM_EMB_V, nnz_g, stream);
  run_spmm(idx_cii, vals_cii, COM_V, COM_EMB_V + UE, nnz_g, stream);
  // t, personal:
  hipMemsetAsync(UTMP, 0, UE * sizeof(float), stream);
  run_spmm(idx_adj, vals_adj, PERS_T, UTMP, nnz_adj, stream);
  scale_rows_k<<<ue_blocks, 256, 0, stream>>>(UTMP, num_inters, UE);
  hipMemsetAsync(PERS_EMB_T, 0, NEb, stream);
  run_spmm(idx_txt_uu, vals_txt_uu, UTMP, PERS_EMB_T, nnz_g, stream);
  run_spmm(idx_txt_ii, vals_txt_ii, PERS_T, PERS_EMB_T + UE, nnz_g, stream);
  // t, common:
  hipMemsetAsync(UTMP, 0, UE * sizeof(float), stream);
  run_spmm(idx_adj, vals_adj, COM_T, UTMP, nnz_adj, stream);
  scale_rows_k<<<ue_blocks, 256, 0, stream>>>(UTMP, num_inters, UE);
  hipMemsetAsync(COM_EMB_T, 0, NEb, stream);
  run_spmm(idx_cuu, vals_cuu, UTMP, COM_EMB_T, nnz_g, stream);
  run_spmm(idx_cii, vals_cii, COM_T, COM_EMB_T + UE, nnz_g, stream);

  // ---- Final attention mix + L2 norms + combine ----
  final_combine_k<<<CDIV(N_NODES, 8), 256, 0, stream>>>(
      CGE, PERS_EMB_V, PERS_EMB_T, COM_EMB_V, COM_EMB_T, att, (float*)d_out,
      N_NODES);
}